// TemporalCrosscoder_11570641895504
// MI455X (gfx1250) — compile-verified
//
#include <hip/hip_runtime.h>
#include <hip/hip_bf16.h>

typedef __attribute__((ext_vector_type(2))) float v2f;
typedef __attribute__((ext_vector_type(8))) float v8f;

#define B_SZ   1024
#define T_SZ   4
#define D_IN   768
#define KDIM   (T_SZ * D_IN)     // 3072 contraction dim
#define DSAE   8192
#define KSEL   32

// ---------------- Encode GEMM:  pre = x(1024x3072) * W(3072x8192) + b_enc ----
// Block tile 64(M) x 128(N), K-step 32. 8 waves: 4(M) x 2(N), each wave 16x64.
#define BM   64
#define BN   128
#define BK   32
#define LDA  36    // A row stride (dwords): mult of 4, conflict-free frag reads
#define LDBP 288   // B pair-row stride (dwords): 2*BN+32 -> half-waves bank-disjoint

__global__ __launch_bounds__(256) void enc_gemm_kernel(
    const float* __restrict__ x, const float* __restrict__ W,
    const float* __restrict__ b_enc, float* __restrict__ pre)
{
    __shared__ float As[BM * LDA];          // 9.2 KB
    __shared__ float Bs2[(BK / 2) * LDBP];  // 18.4 KB, K-pair interleaved

    const int tid  = threadIdx.x;
    const int lane = tid & 31;
    const int wave = tid >> 5;       // 0..7
    const int wm   = wave & 3;       // M sub-tile 0..3
    const int wn   = wave >> 2;      // N sub-tile 0..1
    const int lrow = lane & 15;      // M (A) / N (B) index within fragment
    const int lhk  = lane >> 4;      // half-wave: K offset selector

    const int bm0 = blockIdx.y * BM;
    const int n0  = blockIdx.x * BN;

    v8f acc[4] = {};

    for (int k0 = 0; k0 < KDIM; k0 += BK) {
        // A tile: 64 rows x 32 k -> 512 float4, 2 per thread (coalesced)
        #pragma unroll
        for (int i = 0; i < 2; ++i) {
            int idx = tid + i * 256;
            int row = idx >> 3;
            int kq  = idx & 7;
            float4 v = *(const float4*)(x + (size_t)(bm0 + row) * KDIM + k0 + kq * 4);
            *(float4*)(&As[row * LDA + kq * 4]) = v;
        }
        // B tile: 32 rows x 128 cols, stored K-pair interleaved:
        //   Bs2[(k>>1)*LDBP + col*2 + (k&1)]
        // so each lane's (K,K+1) fragment pair is one aligned b64 LDS load.
        #pragma unroll
        for (int i = 0; i < 4; ++i) {
            int idx = tid + i * 256;       // 0..1023
            int pr  = idx >> 6;            // pair-row 0..15
            int cp  = idx & 63;            // col-pair 0..63
            int col = cp * 2;
            const float* g0 = W + (size_t)(k0 + 2 * pr    ) * DSAE + n0 + col;
            const float* g1 = W + (size_t)(k0 + 2 * pr + 1) * DSAE + n0 + col;
            float2 r0 = *(const float2*)g0;
            float2 r1 = *(const float2*)g1;
            float4 v = make_float4(r0.x, r1.x, r0.y, r1.y);
            *(float4*)(&Bs2[pr * LDBP + col * 2]) = v;
        }
        __syncthreads();

        #pragma unroll
        for (int kk = 0; kk < BK; kk += 4) {
            // A fragment 16x4 f32: lane=M, vgpr v holds K = kk + 2*lhk + v
            v2f a = *(const v2f*)(&As[(wm * 16 + lrow) * LDA + kk + 2 * lhk]);
            const int prb = (kk >> 1) + lhk;   // pair-row for K = kk + 2*lhk
            #pragma unroll
            for (int t = 0; t < 4; ++t) {
                // B fragment 4x16 f32: lane=N, vgpr v holds K = kk + 2*lhk + v
                const int col = wn * 64 + t * 16 + lrow;
                v2f b = *(const v2f*)(&Bs2[prb * LDBP + col * 2]);
                acc[t] = __builtin_amdgcn_wmma_f32_16x16x4_f32(
                    false, a, false, b, (short)0, acc[t], false, false);
            }
        }
        __syncthreads();
    }

    // Epilogue: C/D layout — vgpr r, lanes0-15: M=r, lanes16-31: M=r+8; N=lrow
    #pragma unroll
    for (int t = 0; t < 4; ++t) {
        int col = n0 + wn * 64 + t * 16 + lrow;
        float bias = b_enc[col];
        #pragma unroll
        for (int r = 0; r < 8; ++r) {
            int row = bm0 + wm * 16 + r + 8 * lhk;
            pre[(size_t)row * DSAE + col] = acc[t][r] + bias;
        }
    }
}

// ---------------- TopK(32) per row, ReLU, deterministic tie-break ------------
__global__ __launch_bounds__(256) void topk_kernel(
    float* __restrict__ z, int* __restrict__ selIdx, float* __restrict__ selVal)
{
    __shared__ unsigned keys[DSAE];      // 32 KB
    __shared__ int s_cnt;
    __shared__ int s_gt[256];
    __shared__ int s_tie[256];

    const int t   = threadIdx.x;
    const int row = blockIdx.x;
    float* zr = z + (size_t)row * DSAE;

    // Load interleaved (coalesced), map float -> order-preserving uint key
    for (int j = 0; j < 32; ++j) {
        int i = j * 256 + t;
        unsigned b = __float_as_uint(zr[i]);
        keys[i] = (b & 0x80000000u) ? ~b : (b | 0x80000000u);
    }
    __syncthreads();

    // Binary search for the K-th largest key: max thresh with count(>=thresh)>=K
    unsigned lo = 0u, hi = 0xFFFFFFFFu;
    while (lo < hi) {
        unsigned mid = lo + ((hi - lo) >> 1) + 1u;   // mid in (lo, hi]
        if (t == 0) s_cnt = 0;
        __syncthreads();
        int c = 0;
        for (int j = 0; j < 32; ++j)
            c += (keys[j * 256 + t] >= mid) ? 1 : 0;
        c += __shfl_xor(c, 1);  c += __shfl_xor(c, 2);  c += __shfl_xor(c, 4);
        c += __shfl_xor(c, 8);  c += __shfl_xor(c, 16);
        if ((t & 31) == 0) atomicAdd(&s_cnt, c);
        __syncthreads();
        int cnt = s_cnt;
        __syncthreads();
        if (cnt >= KSEL) lo = mid; else hi = mid - 1u;
    }
    const unsigned thresh = lo;

    // count strictly greater than thresh
    if (t == 0) s_cnt = 0;
    __syncthreads();
    {
        int c = 0;
        for (int j = 0; j < 32; ++j)
            c += (keys[j * 256 + t] > thresh) ? 1 : 0;
        c += __shfl_xor(c, 1);  c += __shfl_xor(c, 2);  c += __shfl_xor(c, 4);
        c += __shfl_xor(c, 8);  c += __shfl_xor(c, 16);
        if ((t & 31) == 0) atomicAdd(&s_cnt, c);
    }
    __syncthreads();
    const int count_gt = s_cnt;
    const int need = KSEL - count_gt;       // ties taken in index order
    __syncthreads();

    // Contiguous-chunk pass: per-thread counts -> serial exclusive prefix
    const int base = t * 32;
    {
        int gtc = 0, tiec = 0;
        for (int j = 0; j < 32; ++j) {
            unsigned kv = keys[base + j];
            gtc  += (kv > thresh) ? 1 : 0;
            tiec += (kv == thresh) ? 1 : 0;
        }
        s_gt[t] = gtc;
        s_tie[t] = tiec;
    }
    __syncthreads();
    if (t == 0) {
        int ag = 0, at_ = 0;
        for (int i = 0; i < 256; ++i) {
            int g = s_gt[i], ti = s_tie[i];
            s_gt[i] = ag;  s_tie[i] = at_;
            ag += g;  at_ += ti;
        }
    }
    __syncthreads();

    int g = s_gt[t];
    int e = s_tie[t];
    int* sI   = selIdx + row * KSEL;
    float* sV = selVal + row * KSEL;
    for (int j = 0; j < 32; ++j) {
        int i = base + j;
        unsigned kv = keys[i];
        float relu = (kv >= 0x80000000u) ? __uint_as_float(kv & 0x7FFFFFFFu) : 0.0f;
        float outv = 0.0f;
        if (kv > thresh) {
            sI[g] = i;  sV[g] = relu;  ++g;
            outv = relu;
        } else if (kv == thresh) {
            if (e < need) {
                int slot = count_gt + e;
                sI[slot] = i;  sV[slot] = relu;
                outv = relu;
            }
            ++e;
        }
        zr[i] = outv;
    }
}

// ---------------- Sparse decode + per-sample squared-error partial -----------
__global__ __launch_bounds__(256) void decode_kernel(
    const float* __restrict__ x, const float* __restrict__ Wd,
    const float* __restrict__ bdec, const int* __restrict__ selIdx,
    const float* __restrict__ selVal, float* __restrict__ xhat,
    float* __restrict__ partial)
{
    __shared__ int   sIdx[KSEL];
    __shared__ float sVal[KSEL];
    __shared__ float red[256];

    const int t = threadIdx.x;
    const int b = blockIdx.x;
    if (t < KSEL) {
        sIdx[t] = selIdx[b * KSEL + t];
        sVal[t] = selVal[b * KSEL + t];
    }
    __syncthreads();

    const float* xr = x    + (size_t)b * KDIM;
    float*       xo = xhat + (size_t)b * KDIM;
    float sq = 0.0f;
    #pragma unroll
    for (int j = 0; j < KDIM / 256; ++j) {
        int pos = j * 256 + t;
        float acc = bdec[pos];
        #pragma unroll 8
        for (int a = 0; a < KSEL; ++a)
            acc = fmaf(sVal[a], Wd[(size_t)sIdx[a] * KDIM + pos], acc);
        xo[pos] = acc;
        float d = acc - xr[pos];
        sq = fmaf(d, d, sq);
    }

    red[t] = sq;
    __syncthreads();
    for (int s = 128; s > 0; s >>= 1) {
        if (t < s) red[t] += red[t + s];
        __syncthreads();
    }
    if (t == 0) partial[b] = red[0];
}

// ---------------- Final deterministic loss reduction -------------------------
__global__ __launch_bounds__(256) void loss_kernel(
    const float* __restrict__ partial, float* __restrict__ loss)
{
    __shared__ float red[256];
    const int t = threadIdx.x;
    float s = (partial[t] + partial[t + 256]) + (partial[t + 512] + partial[t + 768]);
    red[t] = s;
    __syncthreads();
    for (int st = 128; st > 0; st >>= 1) {
        if (t < st) red[t] += red[t + st];
        __syncthreads();
    }
    if (t == 0) *loss = red[0] * (1.0f / (float)(B_SZ * T_SZ));
}

extern "C" void kernel_launch(void* const* d_in, const int* in_sizes, int n_in,
                              void* d_out, int out_size, void* d_ws, size_t ws_size,
                              hipStream_t stream)
{
    (void)in_sizes; (void)n_in; (void)out_size; (void)ws_size;
    const float* x     = (const float*)d_in[0];
    const float* W_enc = (const float*)d_in[1];
    const float* W_dec = (const float*)d_in[2];
    const float* b_enc = (const float*)d_in[3];
    const float* b_dec = (const float*)d_in[4];
    // d_in[5] is k == 32, fixed at compile time (KSEL)

    float* out  = (float*)d_out;
    float* loss = out;                                   // [0]
    float* xhat = out + 1;                               // [1 .. 1+B*T*D_IN)
    float* z    = out + 1 + (size_t)B_SZ * T_SZ * D_IN;  // pre, then sparse z

    int*   selIdx  = (int*)d_ws;                         // B*K ints
    float* selVal  = (float*)d_ws + (size_t)B_SZ * KSEL; // B*K floats
    float* partial = (float*)d_ws + (size_t)2 * B_SZ * KSEL; // B floats

    dim3 ggrid(DSAE / BN, B_SZ / BM);   // 64 x 16
    enc_gemm_kernel<<<ggrid, 256, 0, stream>>>(x, W_enc, b_enc, z);
    topk_kernel<<<B_SZ, 256, 0, stream>>>(z, selIdx, selVal);
    decode_kernel<<<B_SZ, 256, 0, stream>>>(x, W_dec, b_dec, selIdx, selVal, xhat, partial);
    loss_kernel<<<1, 256, 0, stream>>>(partial, loss);
}